// LSTMDecoder_22900765622283
// MI455X (gfx1250) — compile-verified
//
#include <hip/hip_runtime.h>
#include <cstddef>

// LSTM decoder: T=512, B=64, I=256, H=512, L=2, fp32 throughout.
// Fused per-timestep kernel: g = x_t @ Wih^T + h @ Whh^T (fp32 WMMA 16x16x4),
// then gate nonlinearities + state update. 1024 sequential launches (graph-replayed).

#define T_ 512
#define B_ 64
#define I_ 256
#define H_ 512

typedef __attribute__((ext_vector_type(2))) float v2f;
typedef __attribute__((ext_vector_type(8))) float v8f;

__device__ __forceinline__ float sigf(float x) { return 1.0f / (1.0f + expf(-x)); }

// Grid: 32 blocks (16 hidden units each). Block: 256 threads = 8 waves.
// Wave w: gate g = w>>1 (columns g*512 + j0 .. +15), batch half mh = w&1 (rows mh*32 .. +31).
__global__ __launch_bounds__(256) void lstm_step_kernel(
    const float* __restrict__ xt, int Kx,
    const float* __restrict__ w_ih, const float* __restrict__ w_hh,
    const float* __restrict__ b_ih, const float* __restrict__ b_hh,
    const float* __restrict__ h_in, float* __restrict__ h_out,
    float* __restrict__ c_st, float* __restrict__ hs_out)
{
    __shared__ float gbuf[64][65];   // [batch][4 gates * 16 cols], padded

    const int tid   = threadIdx.x;
    const int wave  = tid >> 5;
    const int lane  = tid & 31;
    const int r     = lane & 15;     // row/col within 16
    const int hi    = lane >> 4;     // 0: K pair {0,1}, 1: K pair {2,3}
    const int ksel2 = hi << 1;
    const int g     = wave >> 1;     // gate 0..3 (i,f,g,o)
    const int mh    = wave & 1;      // batch half
    const int j0    = blockIdx.x << 4;       // hidden slice base
    const int n0    = g * H_ + j0;           // W row base for this wave's 16 cols

    const int mb0 = mh * 32;
    const int mb1 = mb0 + 16;

    v8f acc0 = {};
    v8f acc1 = {};

    // ---- input-side contribution: x_t[B,Kx] @ w_ih[N,Kx]^T ----
    {
        const float* pa0 = xt + (size_t)(mb0 + r) * Kx + ksel2;
        const float* pa1 = xt + (size_t)(mb1 + r) * Kx + ksel2;
        const float* pb  = w_ih + (size_t)(n0 + r) * Kx + ksel2;
        for (int k = 0; k < Kx; k += 4) {
            v2f bf = *(const v2f*)(pb + k);
            v2f a0 = *(const v2f*)(pa0 + k);
            v2f a1 = *(const v2f*)(pa1 + k);
            acc0 = __builtin_amdgcn_wmma_f32_16x16x4_f32(false, a0, false, bf,
                                                         (short)0, acc0, false, false);
            acc1 = __builtin_amdgcn_wmma_f32_16x16x4_f32(false, a1, false, bf,
                                                         (short)0, acc1, false, false);
        }
    }
    // ---- recurrent contribution: h[B,H] @ w_hh[N,H]^T ----
    {
        const float* pa0 = h_in + (size_t)(mb0 + r) * H_ + ksel2;
        const float* pa1 = h_in + (size_t)(mb1 + r) * H_ + ksel2;
        const float* pb  = w_hh + (size_t)(n0 + r) * H_ + ksel2;
        for (int k = 0; k < H_; k += 4) {
            v2f bf = *(const v2f*)(pb + k);
            v2f a0 = *(const v2f*)(pa0 + k);
            v2f a1 = *(const v2f*)(pa1 + k);
            acc0 = __builtin_amdgcn_wmma_f32_16x16x4_f32(false, a0, false, bf,
                                                         (short)0, acc0, false, false);
            acc1 = __builtin_amdgcn_wmma_f32_16x16x4_f32(false, a1, false, bf,
                                                         (short)0, acc1, false, false);
        }
    }

    // C layout: VGPR e -> row e (lanes 0-15) / row e+8 (lanes 16-31), col = lane&15
    const int rowsel = hi << 3;
    const int cL = (g << 4) + r;
#pragma unroll
    for (int e = 0; e < 8; ++e) {
        gbuf[mb0 + rowsel + e][cL] = acc0[e];
        gbuf[mb1 + rowsel + e][cL] = acc1[e];
    }
    __syncthreads();

    // ---- gate nonlinearities + state update: 64 batch x 16 hidden ----
    for (int e = tid; e < B_ * 16; e += 256) {
        const int b  = e >> 4;
        const int jl = e & 15;
        const int j  = j0 + jl;
        float gi = gbuf[b][jl]      + b_ih[j]          + b_hh[j];
        float gf = gbuf[b][16 + jl] + b_ih[H_ + j]     + b_hh[H_ + j];
        float gc = gbuf[b][32 + jl] + b_ih[2 * H_ + j] + b_hh[2 * H_ + j];
        float go = gbuf[b][48 + jl] + b_ih[3 * H_ + j] + b_hh[3 * H_ + j];
        const size_t idx = (size_t)b * H_ + j;
        float cn = sigf(gf) * c_st[idx] + sigf(gi) * tanhf(gc);
        float hn = sigf(go) * tanhf(cn);
        c_st[idx]  = cn;
        h_out[idx] = hn;
        if (hs_out) hs_out[idx] = hn;   // hs_out pre-offset by t*B*H
    }
}

extern "C" void kernel_launch(void* const* d_in, const int* in_sizes, int n_in,
                              void* d_out, int out_size, void* d_ws, size_t ws_size,
                              hipStream_t stream) {
    (void)in_sizes; (void)n_in; (void)out_size; (void)ws_size;

    const float* x    = (const float*)d_in[0];
    const float* h0   = (const float*)d_in[1];
    const float* c0   = (const float*)d_in[2];
    const float* wih0 = (const float*)d_in[3];
    const float* whh0 = (const float*)d_in[4];
    const float* bih0 = (const float*)d_in[5];
    const float* bhh0 = (const float*)d_in[6];
    const float* wih1 = (const float*)d_in[7];
    const float* whh1 = (const float*)d_in[8];
    const float* bih1 = (const float*)d_in[9];
    const float* bhh1 = (const float*)d_in[10];
    float* out = (float*)d_out;

    // workspace carve: hs0[T*B*H] | hA[B*H] | hB[B*H] | c[B*H]   (~65 MB)
    float* hs0 = (float*)d_ws;
    float* hA  = hs0 + (size_t)T_ * B_ * H_;
    float* hB  = hA + (size_t)B_ * H_;
    float* cb  = hB + (size_t)B_ * H_;

    const size_t bh = (size_t)B_ * H_;
    const size_t bh_bytes = bh * sizeof(float);

    // ---------------- layer 0 ----------------
    hipMemcpyAsync(hA, h0, bh_bytes, hipMemcpyDeviceToDevice, stream);
    hipMemcpyAsync(cb, c0, bh_bytes, hipMemcpyDeviceToDevice, stream);
    for (int t = 0; t < T_; ++t) {
        const float* hin = (t & 1) ? hB : hA;
        float*       hout = (t & 1) ? hA : hB;
        lstm_step_kernel<<<dim3(32), dim3(256), 0, stream>>>(
            x + (size_t)t * B_ * I_, I_, wih0, whh0, bih0, bhh0,
            hin, hout, cb, hs0 + (size_t)t * bh);
    }
    // t=511 (odd) wrote hA
    hipMemcpyAsync(out,          hA, bh_bytes, hipMemcpyDeviceToDevice, stream); // h_n[0]
    hipMemcpyAsync(out + 2 * bh, cb, bh_bytes, hipMemcpyDeviceToDevice, stream); // c_n[0]

    // ---------------- layer 1 ----------------
    hipMemcpyAsync(hA, h0 + bh, bh_bytes, hipMemcpyDeviceToDevice, stream);
    hipMemcpyAsync(cb, c0 + bh, bh_bytes, hipMemcpyDeviceToDevice, stream);
    for (int t = 0; t < T_; ++t) {
        const float* hin = (t & 1) ? hB : hA;
        float*       hout = (t & 1) ? hA : hB;
        lstm_step_kernel<<<dim3(32), dim3(256), 0, stream>>>(
            hs0 + (size_t)t * bh, H_, wih1, whh1, bih1, bhh1,
            hin, hout, cb, nullptr);
    }
    hipMemcpyAsync(out + bh,     hA, bh_bytes, hipMemcpyDeviceToDevice, stream); // h_n[1]
    hipMemcpyAsync(out + 3 * bh, cb, bh_bytes, hipMemcpyDeviceToDevice, stream); // c_n[1]

    // inputs passthrough
    hipMemcpyAsync(out + 4 * bh, x, (size_t)T_ * B_ * I_ * sizeof(float),
                   hipMemcpyDeviceToDevice, stream);
}